// TransformerConv2d_35493609734786
// MI455X (gfx1250) — compile-verified
//
#include <hip/hip_runtime.h>
#include <math.h>

typedef __attribute__((ext_vector_type(2))) float v2f;
typedef __attribute__((ext_vector_type(8))) float v8f;

#define BB 4
#define CC 64
#define HIN 5
#define WIN 5
#define KS 3
#define EE 12
#define NHEAD 4
#define DHD 3
#define FFD 24
#define HO 3
#define WO 3
#define LL 9
#define SS 2304              /* B*C*K*K */
#define ROWS (SS*LL)         /* 20736 */
#define STILES (SS/16)       /* 144 */
#define NHTOT (LL*NHEAD)     /* 36 */

/* workspace offsets in floats (all multiples of 4 -> float4 aligned) */
#define OFF_SEQ  0
#define OFF_Q    248832
#define OFF_K    580608
#define OFF_V    912384
#define OFF_CTX  1244160
#define OFF_PROJ 1492992

/* ---------------- stage A: im2col + emb + qkv projection ---------------- */
__global__ void build_seq_qkv(const float* __restrict__ x,
                              const float* __restrict__ ch_emb,
                              const float* __restrict__ pos_emb,
                              const float* __restrict__ w_in,   /* (36,12) */
                              const float* __restrict__ b_in,   /* (36,)   */
                              float* __restrict__ ws)
{
    int t = blockIdx.x * blockDim.x + threadIdx.x;
    if (t >= ROWS) return;
    int s  = t / LL, n = t % LL;
    int b  = s / (CC * KS * KS);
    int rem = s % (CC * KS * KS);
    int c  = rem / (KS * KS);
    int kk = rem % (KS * KS);
    int i = kk / KS, j = kk % KS;
    int ho = n / WO, wo = n % WO;
    float xv = x[((b * CC + c) * HIN + (i + ho)) * WIN + (j + wo)];

    float seq[EE];
    float* seqout = ws + OFF_SEQ + (size_t)t * EE;
#pragma unroll
    for (int e = 0; e < EE; ++e) {
        seq[e] = ch_emb[c * EE + e] + pos_emb[n * EE + e] + xv;
        seqout[e] = seq[e];
    }

    float qkv[36];
#pragma unroll
    for (int eo = 0; eo < 36; ++eo) {
        float acc = b_in[eo];
        const float* wr = w_in + eo * EE;
#pragma unroll
        for (int e = 0; e < EE; ++e) acc = fmaf(seq[e], wr[e], acc);
        qkv[eo] = acc;
    }

    const float rs3 = 0.57735026919f; /* 1/sqrt(DH) folded into Q */
    float4* Q4 = (float4*)(ws + OFF_Q);
    float4* K4 = (float4*)(ws + OFF_K);
    float4* V4 = (float4*)(ws + OFF_V);
#pragma unroll
    for (int h = 0; h < NHEAD; ++h) {
        int nh = n * NHEAD + h;
        Q4[(size_t)nh * SS + s] = make_float4(qkv[h*3+0]*rs3, qkv[h*3+1]*rs3, qkv[h*3+2]*rs3, 0.f);
        K4[(size_t)nh * SS + s] = make_float4(qkv[12+h*3+0], qkv[12+h*3+1], qkv[12+h*3+2], 0.f);
        V4[(size_t)nh * SS + s] = make_float4(qkv[24+h*3+0], qkv[24+h*3+1], qkv[24+h*3+2], 0.f);
    }
}

/* ------- stage B: flash attention, WMMA f32 16x16x4, lazy softmax ------- */
/* Each lane keeps a LANE-LOCAL running max and partial sums normalized by
   exp(-m_lane) over the t-columns it owns.  No cross-lane traffic in the
   144-iteration loop; one butterfly reduction per row in the epilogue.    */
__global__ void __launch_bounds__(32) flash_attn(float* __restrict__ ws)
{
    const int lane = threadIdx.x;
    const int hi   = lane >> 4;          /* half-wave id */
    const int l15  = lane & 15;
    const int sTile = blockIdx.x;
    const int nh    = blockIdx.y;

    const float4* Q4 = (const float4*)(ws + OFF_Q) + (size_t)nh * SS;
    const float4* K4 = (const float4*)(ws + OFF_K) + (size_t)nh * SS;
    const float4* V4 = (const float4*)(ws + OFF_V) + (size_t)nh * SS;

    const int s0 = sTile * 16;
    /* A-matrix (16x4 f32): lanes 0-15 hold K=0/1, lanes 16-31 hold K=2/3 */
    float4 q4 = Q4[s0 + l15];
    v2f a;
    a.x = hi ? q4.z : q4.x;
    a.y = hi ? q4.w : q4.y;   /* q4.w == 0 (pad) */

    float m[8], lsum[8], o0[8], o1[8], o2[8];
#pragma unroll
    for (int r = 0; r < 8; ++r) { m[r] = -3.0e38f; lsum[r] = 0.f; o0[r] = 0.f; o1[r] = 0.f; o2[r] = 0.f; }

    for (int tt = 0; tt < STILES; ++tt) {
        const int t0 = tt * 16;
        float4 k4 = K4[t0 + l15];
        float4 v4 = V4[t0 + l15];
        /* B-matrix (4x16 f32): lanes 0-15 rows K=0/1, lanes 16-31 rows K=2/3 */
        v2f bm;
        bm.x = hi ? k4.z : k4.x;
        bm.y = hi ? k4.w : k4.y;
        v8f c = {0.f, 0.f, 0.f, 0.f, 0.f, 0.f, 0.f, 0.f};
        /* scores tile: D[r] = row s0+r (lanes 0-15) / s0+r+8 (lanes 16-31),
           column t0 + (lane&15) */
        v8f d = __builtin_amdgcn_wmma_f32_16x16x4_f32(
            false, a, false, bm, (short)0, c, false, false);

#pragma unroll
        for (int r = 0; r < 8; ++r) {
            float xv   = d[r];
            float mnew = fmaxf(m[r], xv);            /* lane-local max  */
            float sc   = __expf(m[r] - mnew);        /* ==1 if no change */
            float p    = __expf(xv - mnew);
            m[r] = mnew;
            lsum[r] = fmaf(lsum[r], sc, p);
            o0[r]   = fmaf(o0[r],   sc, p * v4.x);
            o1[r]   = fmaf(o1[r],   sc, p * v4.y);
            o2[r]   = fmaf(o2[r],   sc, p * v4.z);
        }
    }

    /* epilogue: single cross-lane reduction per row (16 lanes per half) */
#pragma unroll
    for (int r = 0; r < 8; ++r) {
        /* row max across lanes */
        float M = m[r];
        M = fmaxf(M, __shfl_xor(M, 1, 32));
        M = fmaxf(M, __shfl_xor(M, 2, 32));
        M = fmaxf(M, __shfl_xor(M, 4, 32));
        M = fmaxf(M, __shfl_xor(M, 8, 32));
        /* rescale this lane's partials to the common max */
        float f = __expf(m[r] - M);
        lsum[r] *= f;  o0[r] *= f;  o1[r] *= f;  o2[r] *= f;
        /* sum across lanes */
#pragma unroll
        for (int msk = 1; msk <= 8; msk <<= 1) {
            lsum[r] += __shfl_xor(lsum[r], msk, 32);
            o0[r]   += __shfl_xor(o0[r],   msk, 32);
            o1[r]   += __shfl_xor(o1[r],   msk, 32);
            o2[r]   += __shfl_xor(o2[r],   msk, 32);
        }
    }

    if (l15 == 0) {  /* lanes 0 and 16 write their 8 rows each */
        const int n = nh >> 2, h = nh & 3;
#pragma unroll
        for (int r = 0; r < 8; ++r) {
            int s = s0 + r + 8 * hi;
            float inv = 1.0f / lsum[r];
            float* ctx = ws + OFF_CTX + ((size_t)s * LL + n) * EE + h * DHD;
            ctx[0] = o0[r] * inv;
            ctx[1] = o1[r] * inv;
            ctx[2] = o2[r] * inv;
        }
    }
}

/* ---------------- stage C: out_proj + LN + FFN + LN + linear ------------ */
__device__ inline void ln12(float* v, const float* w, const float* b)
{
    float mu = 0.f;
#pragma unroll
    for (int e = 0; e < EE; ++e) mu += v[e];
    mu *= (1.0f / EE);
    float var = 0.f;
#pragma unroll
    for (int e = 0; e < EE; ++e) { float d = v[e] - mu; var = fmaf(d, d, var); }
    var *= (1.0f / EE);
    float inv = rsqrtf(var + 1e-5f);
#pragma unroll
    for (int e = 0; e < EE; ++e) v[e] = (v[e] - mu) * inv * w[e] + b[e];
}

__global__ void mlp_rows(float* __restrict__ ws,
                         const float* __restrict__ ow,  const float* __restrict__ ob,
                         const float* __restrict__ l1w, const float* __restrict__ l1b,
                         const float* __restrict__ f1w, const float* __restrict__ f1b,
                         const float* __restrict__ f2w, const float* __restrict__ f2b,
                         const float* __restrict__ l2w, const float* __restrict__ l2b,
                         const float* __restrict__ lw,  const float* __restrict__ lb)
{
    int t = blockIdx.x * blockDim.x + threadIdx.x;
    if (t >= ROWS) return;
    const float* seqr = ws + OFF_SEQ + (size_t)t * EE;
    const float* ctxr = ws + OFF_CTX + (size_t)t * EE;

    float h1[EE];
#pragma unroll
    for (int e = 0; e < EE; ++e) {
        float acc = ob[e];
        const float* wr = ow + e * EE;
#pragma unroll
        for (int e2 = 0; e2 < EE; ++e2) acc = fmaf(ctxr[e2], wr[e2], acc);
        h1[e] = seqr[e] + acc;
    }
    ln12(h1, l1w, l1b);

    float ffh[FFD];
#pragma unroll
    for (int f = 0; f < FFD; ++f) {
        float acc = f1b[f];
        const float* wr = f1w + f * EE;
#pragma unroll
        for (int e = 0; e < EE; ++e) acc = fmaf(h1[e], wr[e], acc);
        ffh[f] = fmaxf(acc, 0.f);
    }

    float h2[EE];
#pragma unroll
    for (int e = 0; e < EE; ++e) {
        float acc = f2b[e];
        const float* wr = f2w + e * FFD;
#pragma unroll
        for (int f = 0; f < FFD; ++f) acc = fmaf(ffh[f], wr[f], acc);
        h2[e] = h1[e] + acc;
    }
    ln12(h2, l2w, l2b);

    float pr = lb[0];
#pragma unroll
    for (int e = 0; e < EE; ++e) pr = fmaf(h2[e], lw[e], pr);
    ws[OFF_PROJ + t] = pr;   /* layout (B, C*KK, L) flattened == t = s*L + n */
}

/* ---------------- stage D: fold back to (B,C,H,W) ----------------------- */
__global__ void fold_out(const float* __restrict__ ws, float* __restrict__ out)
{
    int t = blockIdx.x * blockDim.x + threadIdx.x;
    if (t >= BB * CC * HIN * WIN) return;
    int wwi = t % WIN;
    int hhi = (t / WIN) % HIN;
    int c   = (t / (WIN * HIN)) % CC;
    int b   = t / (WIN * HIN * CC);
    const float* proj = ws + OFF_PROJ;
    float acc = 0.f;
#pragma unroll
    for (int i = 0; i < KS; ++i) {
        int ho = hhi - i;
        if (ho < 0 || ho >= HO) continue;
#pragma unroll
        for (int j = 0; j < KS; ++j) {
            int wo = wwi - j;
            if (wo < 0 || wo >= WO) continue;
            acc += proj[(size_t)((b * CC + c) * (KS * KS) + i * KS + j) * LL + ho * WO + wo];
        }
    }
    out[t] = acc;
}

extern "C" void kernel_launch(void* const* d_in, const int* in_sizes, int n_in,
                              void* d_out, int out_size, void* d_ws, size_t ws_size,
                              hipStream_t stream)
{
    (void)in_sizes; (void)n_in; (void)out_size; (void)ws_size;
    const float* x   = (const float*)d_in[0];
    const float* ce  = (const float*)d_in[1];
    const float* pe  = (const float*)d_in[2];
    const float* inw = (const float*)d_in[3];
    const float* inb = (const float*)d_in[4];
    const float* ow  = (const float*)d_in[5];
    const float* ob  = (const float*)d_in[6];
    const float* l1w = (const float*)d_in[7];
    const float* l1b = (const float*)d_in[8];
    const float* f1w = (const float*)d_in[9];
    const float* f1b = (const float*)d_in[10];
    const float* f2w = (const float*)d_in[11];
    const float* f2b = (const float*)d_in[12];
    const float* l2w = (const float*)d_in[13];
    const float* l2b = (const float*)d_in[14];
    const float* lw  = (const float*)d_in[15];
    const float* lb  = (const float*)d_in[16];
    float* ws  = (float*)d_ws;
    float* out = (float*)d_out;

    build_seq_qkv<<<(ROWS + 255) / 256, 256, 0, stream>>>(x, ce, pe, inw, inb, ws);
    flash_attn<<<dim3(STILES, NHTOT), 32, 0, stream>>>(ws);
    mlp_rows<<<(ROWS + 255) / 256, 256, 0, stream>>>(ws, ow, ob, l1w, l1b,
                                                     f1w, f1b, f2w, f2b,
                                                     l2w, l2b, lw, lb);
    fold_out<<<(BB * CC * HIN * WIN + 255) / 256, 256, 0, stream>>>(ws, out);
}